// GNNModel_67104569033231
// MI455X (gfx1250) — compile-verified
//
#include <hip/hip_runtime.h>
#include <hip/hip_bf16.h>

typedef float v2f __attribute__((ext_vector_type(2)));
typedef float v8f __attribute__((ext_vector_type(8)));

#define HID 96
#define EPS 1e-5f

// ---------------------------------------------------------------------------
// Degree / normalization
// ---------------------------------------------------------------------------
__global__ void k_init_deg(float* deg, int N) {
    int i = blockIdx.x * blockDim.x + threadIdx.x;
    if (i < N) deg[i] = 1.0f;                      // self-loop contributes 1
}

__global__ void k_count_deg(const long long* __restrict__ dst, float* deg, int E) {
    int e = blockIdx.x * blockDim.x + threadIdx.x;
    if (e < E) atomicAdd(&deg[(int)dst[e]], 1.0f);
}

__global__ void k_dinv(const float* __restrict__ deg, float* dinv, int N) {
    int i = blockIdx.x * blockDim.x + threadIdx.x;
    if (i < N) dinv[i] = __frsqrt_rn(deg[i]);      // deg >= 1 always
}

// ---------------------------------------------------------------------------
// WMMA f32 GEMM:  H[N,Fout] = X[N,FIN] @ W[FIN,Fout]
// grid.x = N/16 row tiles; block = (Fout/16) waves, wave w -> column tile w.
// A 16x4 layout: lanes 0-15 hold K={k0,k0+1}, lanes 16-31 hold K={k0+2,k0+3}.
// B 4x16 layout: N striped across lanes; K across VGPRs / lane halves.
// D 16x16: VGPR r, lanes 0-15 -> M=r, lanes 16-31 -> M=r+8.
// ---------------------------------------------------------------------------
template <int FIN>
__global__ void k_gemm_wmma(const float* __restrict__ X, const float* __restrict__ W,
                            float* __restrict__ Hout, int Fout) {
    const int wave = threadIdx.x >> 5;
    const int lane = threadIdx.x & 31;
    const int m    = lane & 15;        // row within tile (A) / col within tile (B,D)
    const int kh   = lane >> 4;        // K-half select

    const int rowBase = blockIdx.x * 16;
    const int colBase = wave * 16;

    const float* xrow = X + (size_t)(rowBase + m) * FIN;

    v8f acc = {};
#pragma unroll
    for (int k0 = 0; k0 < FIN; k0 += 4) {
        const int ka = k0 + 2 * kh;
        v2f a, b;
        a.x = xrow[ka];
        a.y = xrow[ka + 1];
        b.x = W[(size_t)ka * Fout + colBase + m];
        b.y = W[(size_t)(ka + 1) * Fout + colBase + m];
        acc = __builtin_amdgcn_wmma_f32_16x16x4_f32(
            /*neg_a=*/false, a, /*neg_b=*/false, b,
            /*c_mod=*/(short)0, acc, /*reuse_a=*/false, /*reuse_b=*/false);
    }

    const int col  = colBase + m;
    const int row0 = rowBase + 8 * kh;
#pragma unroll
    for (int r = 0; r < 8; ++r)
        Hout[(size_t)(row0 + r) * Fout + col] = acc[r];
}

// ---------------------------------------------------------------------------
// agg init: self-loop contribution (norm = dinv^2) + bias
// ---------------------------------------------------------------------------
__global__ void k_init_agg(const float* __restrict__ h, const float* __restrict__ dinv,
                           const float* __restrict__ bias, float* __restrict__ agg, int N) {
    size_t i = (size_t)blockIdx.x * blockDim.x + threadIdx.x;
    if (i >= (size_t)N * HID) return;
    int node = (int)(i / HID);
    int c    = (int)(i % HID);
    float dv = dinv[node];
    agg[i] = h[i] * dv * dv + bias[c];
}

__global__ void k_zero_stats(float* stats) {
    if (threadIdx.x < 2 * HID) stats[threadIdx.x] = 0.0f;
}

// ---------------------------------------------------------------------------
// Edge scatter: one wave per edge, 3 channels per lane (96 = 3*32).
// ---------------------------------------------------------------------------
__global__ void k_scatter(const long long* __restrict__ src, const long long* __restrict__ dst,
                          const float* __restrict__ dinv, const float* __restrict__ h,
                          float* __restrict__ agg, int E) {
    int wid  = (int)(((size_t)blockIdx.x * blockDim.x + threadIdx.x) >> 5);
    int lane = threadIdx.x & 31;
    if (wid >= E) return;
    int s = (int)src[wid];
    int d = (int)dst[wid];
    float nrm = dinv[s] * dinv[d];
    const float* hs = h + (size_t)s * HID;
    float* ad = agg + (size_t)d * HID;
#pragma unroll
    for (int c = lane; c < HID; c += 32)
        atomicAdd(&ad[c], hs[c] * nrm);
}

// ---------------------------------------------------------------------------
// BatchNorm: channel-parallel reduction (sum, sumsq), then fused apply+ReLU.
// blockDim = 96 (one thread per channel); rows strided across blocks.
// ---------------------------------------------------------------------------
__global__ void k_bn_reduce(const float* __restrict__ A, float* __restrict__ stats, int N) {
    int c = threadIdx.x;
    float s = 0.0f, q = 0.0f;
    for (int row = blockIdx.x; row < N; row += gridDim.x) {
        float v = A[(size_t)row * HID + c];
        s += v;
        q += v * v;
    }
    atomicAdd(&stats[c], s);
    atomicAdd(&stats[HID + c], q);
}

__global__ void k_bn_apply_relu(float* __restrict__ A, const float* __restrict__ stats,
                                const float* __restrict__ g, const float* __restrict__ be,
                                int N) {
    size_t i = (size_t)blockIdx.x * blockDim.x + threadIdx.x;
    if (i >= (size_t)N * HID) return;
    int c = (int)(i % HID);
    float invN = 1.0f / (float)N;
    float mean = stats[c] * invN;
    float var  = stats[HID + c] * invN - mean * mean;
    float w    = __frsqrt_rn(var + EPS);
    float y    = (A[i] - mean) * w * g[c] + be[c];
    A[i] = fmaxf(y, 0.0f);
}

// ---------------------------------------------------------------------------
// Final layer (96 -> 2): plain dot products + 2-channel scatter into d_out.
// ---------------------------------------------------------------------------
__global__ void k_out_transform(const float* __restrict__ A, const float* __restrict__ W4,
                                float* __restrict__ h4, int N) {
    int i = blockIdx.x * blockDim.x + threadIdx.x;
    if (i >= N) return;
    const float* row = A + (size_t)i * HID;
    float a0 = 0.0f, a1 = 0.0f;
#pragma unroll
    for (int c = 0; c < HID; ++c) {
        float v = row[c];
        a0 += v * W4[2 * c + 0];
        a1 += v * W4[2 * c + 1];
    }
    h4[2 * (size_t)i + 0] = a0;
    h4[2 * (size_t)i + 1] = a1;
}

__global__ void k_out_init(const float* __restrict__ h4, const float* __restrict__ dinv,
                           const float* __restrict__ b4, float* __restrict__ out, int N) {
    int i = blockIdx.x * blockDim.x + threadIdx.x;
    if (i >= 2 * N) return;
    int node = i >> 1;
    int c    = i & 1;
    float dv = dinv[node];
    out[i] = h4[i] * dv * dv + b4[c];
}

__global__ void k_out_scatter(const long long* __restrict__ src, const long long* __restrict__ dst,
                              const float* __restrict__ dinv, const float* __restrict__ h4,
                              float* __restrict__ out, int E) {
    int e = blockIdx.x * blockDim.x + threadIdx.x;
    if (e >= E) return;
    int s = (int)src[e];
    int d = (int)dst[e];
    float nrm = dinv[s] * dinv[d];
    atomicAdd(&out[2 * (size_t)d + 0], h4[2 * (size_t)s + 0] * nrm);
    atomicAdd(&out[2 * (size_t)d + 1], h4[2 * (size_t)s + 1] * nrm);
}

// ---------------------------------------------------------------------------
static inline int cdiv(long long a, long long b) { return (int)((a + b - 1) / b); }

extern "C" void kernel_launch(void* const* d_in, const int* in_sizes, int n_in,
                              void* d_out, int out_size, void* d_ws, size_t ws_size,
                              hipStream_t stream) {
    const float*     x  = (const float*)d_in[0];
    const long long* ei = (const long long*)d_in[1];   // [2, E] int64
    const float* W1 = (const float*)d_in[2];  const float* b1 = (const float*)d_in[3];
    const float* g1 = (const float*)d_in[4];  const float* be1 = (const float*)d_in[5];
    const float* W2 = (const float*)d_in[6];  const float* b2 = (const float*)d_in[7];
    const float* g2 = (const float*)d_in[8];  const float* be2 = (const float*)d_in[9];
    const float* W3 = (const float*)d_in[10]; const float* b3 = (const float*)d_in[11];
    const float* g3 = (const float*)d_in[12]; const float* be3 = (const float*)d_in[13];
    const float* W4 = (const float*)d_in[14]; const float* b4 = (const float*)d_in[15];

    const int N = in_sizes[0] / 32;   // 50000 (multiple of 16)
    const int E = in_sizes[1] / 2;    // 800000
    const long long* src = ei;
    const long long* dst = ei + E;

    float* ws    = (float*)d_ws;
    float* deg   = ws;
    float* dinv  = ws + N;
    float* bufH  = ws + 2 * (size_t)N;                 // GEMM output [N, 96]
    float* bufA  = bufH + (size_t)N * HID;             // aggregation/activation [N, 96]
    float* stats = bufA + (size_t)N * HID;             // 2*96 floats
    float* h4    = bufH;                               // reuse bufH for [N, 2] in final layer

    // --- normalization coefficients ---
    k_init_deg<<<cdiv(N, 256), 256, 0, stream>>>(deg, N);
    k_count_deg<<<cdiv(E, 256), 256, 0, stream>>>(dst, deg, E);
    k_dinv<<<cdiv(N, 256), 256, 0, stream>>>(deg, dinv, N);

    const int gemmBlk  = (HID / 16) * 32;              // 6 waves = 192 threads
    const int gemmGrd  = N / 16;                       // 3125 row tiles
    const int nhGrid   = cdiv((long long)N * HID, 256);
    const int edgeGrid = cdiv((long long)E * 32, 256);

    // --- layer 1: 32 -> 96 ---
    k_gemm_wmma<32><<<gemmGrd, gemmBlk, 0, stream>>>(x, W1, bufH, HID);
    k_init_agg<<<nhGrid, 256, 0, stream>>>(bufH, dinv, b1, bufA, N);
    k_zero_stats<<<1, 2 * HID, 0, stream>>>(stats);
    k_scatter<<<edgeGrid, 256, 0, stream>>>(src, dst, dinv, bufH, bufA, E);
    k_bn_reduce<<<512, HID, 0, stream>>>(bufA, stats, N);
    k_bn_apply_relu<<<nhGrid, 256, 0, stream>>>(bufA, stats, g1, be1, N);

    // --- layer 2: 96 -> 96 ---
    k_gemm_wmma<96><<<gemmGrd, gemmBlk, 0, stream>>>(bufA, W2, bufH, HID);
    k_init_agg<<<nhGrid, 256, 0, stream>>>(bufH, dinv, b2, bufA, N);
    k_zero_stats<<<1, 2 * HID, 0, stream>>>(stats);
    k_scatter<<<edgeGrid, 256, 0, stream>>>(src, dst, dinv, bufH, bufA, E);
    k_bn_reduce<<<512, HID, 0, stream>>>(bufA, stats, N);
    k_bn_apply_relu<<<nhGrid, 256, 0, stream>>>(bufA, stats, g2, be2, N);

    // --- layer 3: 96 -> 96 ---
    k_gemm_wmma<96><<<gemmGrd, gemmBlk, 0, stream>>>(bufA, W3, bufH, HID);
    k_init_agg<<<nhGrid, 256, 0, stream>>>(bufH, dinv, b3, bufA, N);
    k_zero_stats<<<1, 2 * HID, 0, stream>>>(stats);
    k_scatter<<<edgeGrid, 256, 0, stream>>>(src, dst, dinv, bufH, bufA, E);
    k_bn_reduce<<<512, HID, 0, stream>>>(bufA, stats, N);
    k_bn_apply_relu<<<nhGrid, 256, 0, stream>>>(bufA, stats, g3, be3, N);

    // --- layer 4: 96 -> 2, aggregate straight into d_out ---
    float* out = (float*)d_out;
    k_out_transform<<<cdiv(N, 256), 256, 0, stream>>>(bufA, W4, h4, N);
    k_out_init<<<cdiv(2 * (long long)N, 256), 256, 0, stream>>>(h4, dinv, b4, out, N);
    k_out_scatter<<<cdiv(E, 256), 256, 0, stream>>>(src, dst, dinv, h4, out, E);
}